// PredictYOLO_v3_85478439125830
// MI455X (gfx1250) — compile-verified
//
#include <hip/hip_runtime.h>
#include <hip/hip_bf16.h>

#define DEVINL __device__ __forceinline__

typedef float v2f __attribute__((ext_vector_type(2)));
typedef float v8f __attribute__((ext_vector_type(8)));

#if defined(__gfx1250__) && __has_builtin(__builtin_amdgcn_wmma_f32_16x16x4_f32)
#define HAVE_WMMA_F32X4 1
#else
#define HAVE_WMMA_F32X4 0
#endif

namespace {
constexpr int   kB        = 32;
constexpr int   kHWA      = 22743;
constexpr int   kC        = 80;
constexpr int   kN        = kB * kHWA;       // 727776
constexpr int   kTopK     = 2048;
constexpr int   kCap      = 4096;            // gather capacity (power of 2 for bitonic)
constexpr int   kBins     = 1024;
constexpr int   kSuppW    = kTopK / 32;      // 64 words per row
constexpr float kThr      = 0.3f;
constexpr float kNms      = 0.5f;

DEVINL float sigf(float x) { return 1.0f / (1.0f + expf(-x)); }
DEVINL float clamp01(float x) { return x < 0.f ? 0.f : (x > 1.f ? 1.f : x); }
DEVINL int binOf(float s) {
  int b = (int)((s - kThr) * ((float)kBins / (1.0f - kThr)));
  return b < 0 ? 0 : (b > kBins - 1 ? kBins - 1 : b);
}
} // namespace

// ---------------------------------------------------------------- init
__global__ void k_init(unsigned* __restrict__ hist, unsigned* __restrict__ cnt,
                       unsigned* __restrict__ supp) {
  int i = blockIdx.x * blockDim.x + threadIdx.x;
  if (i < kTopK * kSuppW) supp[i] = 0u;
  if (i < kBins)          hist[i] = 0u;
  if (i < 16)             cnt[i]  = 0u;
}

// ---------------------------------------------------------------- scores (streaming, memory bound)
// one wave32 per anchor; lanes cover classes c, c+32, c+64 -> coalesced 128B loads
__global__ void k_score(const float* __restrict__ pconf,
                        const float* __restrict__ pcls,
                        float* __restrict__ scores,
                        unsigned* __restrict__ hist) {
  int wave = threadIdx.x >> 5;
  int lane = threadIdx.x & 31;
  int e = blockIdx.x * 8 + wave;
  if (e >= kN) return;
  const float* row = pcls + (long long)e * kC;
  __builtin_prefetch(row + 8 * kC, 0, 0);      // stream hint: 8 anchors ahead
  float a = row[lane];
  float b = row[lane + 32];
  float m = a > b ? a : b;
  if (lane < 16) { float c = row[lane + 64]; m = c > m ? c : m; }
#pragma unroll
  for (int off = 16; off; off >>= 1) {
    float o = __shfl_xor(m, off, 32);
    m = o > m ? o : m;
  }
  if (lane == 0) {
    // sigmoid is monotonic: max(sigmoid(cls)) == sigmoid(max(cls))
    float s = sigf(pconf[e]) * sigf(m);
    scores[e] = s;
    if (s > kThr) atomicAdd(&hist[binOf(s)], 1u);
  }
}

// ---------------------------------------------------------------- cutoff bin (top-down histogram scan)
__global__ void k_cutoff(const unsigned* __restrict__ hist, unsigned* __restrict__ cnt) {
  if (blockIdx.x == 0 && threadIdx.x == 0) {
    unsigned cum = 0; int cutoff = 0;
    for (int b = kBins - 1; b >= 0; --b) {
      cum += hist[b];
      if (cum >= (unsigned)kTopK) { cutoff = b; break; }
    }
    cnt[1] = (unsigned)cutoff;
  }
}

// ---------------------------------------------------------------- gather candidates above cutoff
__global__ void k_gather(const float* __restrict__ scores, unsigned* __restrict__ cnt,
                         unsigned long long* __restrict__ candKeys) {
  int e = blockIdx.x * blockDim.x + threadIdx.x;
  if (e >= kN) return;
  float s = scores[e];
  if (s > kThr && binOf(s) >= (int)cnt[1]) {
    unsigned pos = atomicAdd(&cnt[0], 1u);
    if (pos < (unsigned)kCap) {
      // key: score bits desc, then index asc (positive float bits are monotonic)
      candKeys[pos] = ((unsigned long long)__float_as_uint(s) << 32)
                    | (unsigned long long)(0xFFFFFFFFu - (unsigned)e);
    }
  }
}

// ---------------------------------------------------------------- bitonic sort 4096 keys in LDS (descending)
__global__ __launch_bounds__(1024) void k_sort(const unsigned* __restrict__ cnt,
                       const unsigned long long* __restrict__ candKeys,
                       unsigned long long* __restrict__ topKeys) {
  __shared__ unsigned long long sk[kCap];
  unsigned m = cnt[0]; if (m > (unsigned)kCap) m = (unsigned)kCap;
  for (int n = threadIdx.x; n < kCap; n += 1024)
    sk[n] = (n < (int)m) ? candKeys[n] : 0ull;   // key 0 sorts last
  __syncthreads();
  for (int k = 2; k <= kCap; k <<= 1) {
    for (int j = k >> 1; j > 0; j >>= 1) {
      for (int n = threadIdx.x; n < kCap; n += 1024) {
        int ixj = n ^ j;
        if (ixj > n) {
          bool asc = (n & k) != 0;               // overall descending
          unsigned long long x = sk[n], y = sk[ixj];
          if ((x < y) != asc) { sk[n] = y; sk[ixj] = x; }
        }
      }
      __syncthreads();
    }
  }
  for (int n = threadIdx.x; n < kTopK; n += 1024) topKeys[n] = sk[n];
}

// ---------------------------------------------------------------- decode top-K: box, label, offsets, area
__global__ void k_decode(const float* __restrict__ pcls,
                         const float* __restrict__ ptxywh,
                         const float* __restrict__ grid_xy,
                         const float* __restrict__ anchor_wh,
                         const float* __restrict__ inv_fsize,
                         const unsigned long long* __restrict__ topKeys,
                         float* __restrict__ selBox, float* __restrict__ selBoxOff,
                         float* __restrict__ selArea, int* __restrict__ selLabel,
                         int* __restrict__ selIdb, float* __restrict__ selScore,
                         unsigned* __restrict__ selValid) {
  int wave = threadIdx.x >> 5, lane = threadIdx.x & 31;
  int k = blockIdx.x * 8 + wave;
  if (k >= kTopK) return;
  unsigned long long key = topKeys[k];
  float s = __uint_as_float((unsigned)(key >> 32));
  unsigned e = 0xFFFFFFFFu - (unsigned)(key & 0xFFFFFFFFu);
  int valid = (s > kThr) && (e < (unsigned)kN);
  if (e >= (unsigned)kN) e = 0;

  // argmax over 80 classes (sigmoid monotone -> argmax of raw logits; ties -> lowest c)
  float v = -3.0e38f; int ci = 0x7fffffff;
  const float* row = pcls + (long long)e * kC;
  for (int c = lane; c < kC; c += 32) { float x = row[c]; if (x > v) { v = x; ci = c; } }
#pragma unroll
  for (int off = 16; off; off >>= 1) {
    float ov = __shfl_xor(v, off, 32);
    int   oc = __shfl_xor(ci, off, 32);
    if (ov > v || (ov == v && oc < ci)) { v = ov; ci = oc; }
  }
  if (lane == 0) {
    int label = ci + 1;
    long long e4 = (long long)e * 4;
    float tx = ptxywh[e4 + 0], ty = ptxywh[e4 + 1];
    float tw = ptxywh[e4 + 2], th = ptxywh[e4 + 3];
    int hw = (int)(e % (unsigned)kHWA);
    int bi = (int)(e / (unsigned)kHWA);
    float invf = inv_fsize[hw];
    float cx = (sigf(tx) + grid_xy[hw * 2 + 0]) * invf;
    float cy = (sigf(ty) + grid_xy[hw * 2 + 1]) * invf;
    float w  = expf(tw) * anchor_wh[hw * 2 + 0];
    float h  = expf(th) * anchor_wh[hw * 2 + 1];
    float l  = clamp01(cx - 0.5f * w), t = clamp01(cy - 0.5f * h);
    float r  = clamp01(cx + 0.5f * w), b = clamp01(cy + 0.5f * h);
    float off4 = ((float)bi * (float)(kC + 2) + (float)label) * 4.0f;
    selBox[k * 4 + 0] = l; selBox[k * 4 + 1] = t;
    selBox[k * 4 + 2] = r; selBox[k * 4 + 3] = b;
    selBoxOff[k * 4 + 0] = l + off4; selBoxOff[k * 4 + 1] = t + off4;
    selBoxOff[k * 4 + 2] = r + off4; selBoxOff[k * 4 + 3] = b + off4;
    selArea[k]  = (r - l) * (b - t);   // offset cancels in area
    selLabel[k] = label;
    selIdb[k]   = bi;
    selScore[k] = s;
    selValid[k] = (unsigned)valid;
  }
}

// ---------------------------------------------------------------- suppression matrix: one wave per 16x16 tile
// pairwise (area_i + area_j) via rank-2 WMMA: A=[area_i,1]x{K0,K1}, B={K0=1,K1=area_j}
// IoU test is division-free: inter/(sum-inter+e) > t  <=>  inter*(1+t) > t*(sum+e)
__global__ void k_supp(const float* __restrict__ selBoxOff,
                       const float* __restrict__ selArea,
                       unsigned* __restrict__ supp) {
  int wave = threadIdx.x >> 5, lane = threadIdx.x & 31;
  int t  = blockIdx.x * 8 + wave;          // 0 .. 16383 exactly (no guard: uniform full EXEC)
  int ti = t >> 7, tj = t & 127;
  int i0 = ti * 16, j0 = tj * 16;
  int l15   = lane & 15;
  int halfo = (lane >> 4) * 8;             // D rows v (+0) for lanes 0-15, v+8 for lanes 16-31
  float lowm = (lane < 16) ? 1.0f : 0.0f;  // branchless: keep EXEC all-ones for WMMA
  float ai = selArea[i0 + l15];
  float aj = selArea[j0 + l15];
  v8f sums;
#if HAVE_WMMA_F32X4
  // A (16x4 f32): lanes 0-15 hold row M=lane, VGPR0=K0, VGPR1=K1; lanes 16-31 = K2,K3 (zero)
  v2f A;  A.x  = ai * lowm;  A.y  = lowm;
  // B (4x16 f32): VGPR0 lanes 0-15 = row K0 (ones), VGPR1 lanes 0-15 = row K1 (area_j)
  v2f Bm; Bm.x = lowm;       Bm.y = aj * lowm;
  v8f Cz = {0.f, 0.f, 0.f, 0.f, 0.f, 0.f, 0.f, 0.f};
  sums = __builtin_amdgcn_wmma_f32_16x16x4_f32(
      /*neg_a=*/false, A, /*neg_b=*/false, Bm,
      /*c_mod=*/(short)0, Cz, /*reuse_a=*/false, /*reuse_b=*/false);
#else
#pragma unroll
  for (int vv = 0; vv < 8; ++vv) sums[vv] = selArea[i0 + halfo + vv] + aj;
#endif
  const float4* boxv = reinterpret_cast<const float4*>(selBoxOff);
  int j = j0 + l15;
  float4 bj = boxv[j];
#pragma unroll
  for (int vv = 0; vv < 8; ++vv) {
    int i = i0 + halfo + vv;               // matches 16x16 f32 D VGPR layout
    float4 bi = boxv[i];
    float lx = bi.x > bj.x ? bi.x : bj.x;
    float ly = bi.y > bj.y ? bi.y : bj.y;
    float rx = bi.z < bj.z ? bi.z : bj.z;
    float ry = bi.w < bj.w ? bi.w : bj.w;
    float iw = rx - lx; iw = iw > 0.f ? iw : 0.f;
    float ih = ry - ly; ih = ih > 0.f ? ih : 0.f;
    float inter = iw * ih;
    // inter/(sum-inter+1e-9) > kNms, denominator > 0  ->  division-free compare
    float lhs = inter * (1.0f + kNms);
    float rhs = kNms * (sums[vv] + 1e-9f);
    if (lhs > rhs && j > i)
      atomicOr(&supp[i * kSuppW + (j >> 5)], 1u << (j & 31));
  }
}

// ---------------------------------------------------------------- sequential NMS sweep + output
__global__ void k_nms_out(const unsigned* __restrict__ supp,
                          const unsigned* __restrict__ selValid,
                          const float* __restrict__ selBox,
                          const int* __restrict__ selLabel,
                          const int* __restrict__ selIdb,
                          const float* __restrict__ selScore,
                          float* __restrict__ out) {
  __shared__ unsigned keepW[kSuppW];
  int tid = threadIdx.x;
  if (tid < kSuppW) {
    unsigned w = 0;
    for (int b = 0; b < 32; ++b)
      if (selValid[tid * 32 + b]) w |= (1u << b);
    keepW[tid] = w;
  }
  __syncthreads();
  for (int i = 0; i < kTopK; ++i) {
    bool alive = (keepW[i >> 5] >> (i & 31)) & 1u;
    if (alive && tid < kSuppW) {
      unsigned m = supp[i * kSuppW + tid];
      int wi = i >> 5;
      if (tid < wi)       m = 0u;
      else if (tid == wi) m &= ~((2u << (i & 31)) - 1u);  // keep only bits j>i (wraps to 0 at i%32==31)
      keepW[tid] &= ~m;
    }
    __syncthreads();
  }
  // outputs: [ids_b(2048) | boxes(2048*4) | labels(2048) | scores(2048) | keep(2048)]
  for (int k = tid; k < kTopK; k += 256) {
    bool kp = (keepW[k >> 5] >> (k & 31)) & 1u;
    out[k] = (float)selIdb[k];
    float b0 = selBox[k * 4 + 0], b1 = selBox[k * 4 + 1];
    float b2 = selBox[k * 4 + 2], b3 = selBox[k * 4 + 3];
    out[2048 + 4 * k + 0] = kp ? b0 : 0.0f;
    out[2048 + 4 * k + 1] = kp ? b1 : 0.0f;
    out[2048 + 4 * k + 2] = kp ? b2 : 0.0f;
    out[2048 + 4 * k + 3] = kp ? b3 : 0.0f;
    out[10240 + k] = (float)selLabel[k];
    out[12288 + k] = kp ? selScore[k] : 0.0f;
    out[14336 + k] = kp ? 1.0f : 0.0f;
  }
}

// ---------------------------------------------------------------- launch
extern "C" void kernel_launch(void* const* d_in, const int* in_sizes, int n_in,
                              void* d_out, int out_size, void* d_ws, size_t ws_size,
                              hipStream_t stream) {
  const float* pconf     = (const float*)d_in[0];
  const float* pcls      = (const float*)d_in[1];
  const float* ptxywh    = (const float*)d_in[2];
  const float* grid_xy   = (const float*)d_in[3];
  const float* anchor_wh = (const float*)d_in[4];
  const float* inv_fsize = (const float*)d_in[5];
  float* out = (float*)d_out;

  char* w = (char*)d_ws;
  auto carve = [&](size_t bytes) { char* p = w; w += (bytes + 255) & ~(size_t)255; return p; };
  float*              scores   = (float*)carve((size_t)kN * 4);
  unsigned*           hist     = (unsigned*)carve((size_t)kBins * 4);
  unsigned*           cnt      = (unsigned*)carve(256);
  unsigned long long* candKeys = (unsigned long long*)carve((size_t)kCap * 8);
  unsigned long long* topKeys  = (unsigned long long*)carve((size_t)kTopK * 8);
  float*              selBox    = (float*)carve((size_t)kTopK * 16);
  float*              selBoxOff = (float*)carve((size_t)kTopK * 16);
  float*              selArea   = (float*)carve((size_t)kTopK * 4);
  int*                selLabel  = (int*)carve((size_t)kTopK * 4);
  int*                selIdb    = (int*)carve((size_t)kTopK * 4);
  float*              selScore  = (float*)carve((size_t)kTopK * 4);
  unsigned*           selValid  = (unsigned*)carve((size_t)kTopK * 4);
  unsigned*           supp      = (unsigned*)carve((size_t)kTopK * kSuppW * 4);

  k_init  <<<(kTopK * kSuppW + 255) / 256, 256, 0, stream>>>(hist, cnt, supp);
  k_score <<<(kN + 7) / 8,                 256, 0, stream>>>(pconf, pcls, scores, hist);
  k_cutoff<<<1,                             32, 0, stream>>>(hist, cnt);
  k_gather<<<(kN + 255) / 256,             256, 0, stream>>>(scores, cnt, candKeys);
  k_sort  <<<1,                           1024, 0, stream>>>(cnt, candKeys, topKeys);
  k_decode<<<kTopK / 8,                    256, 0, stream>>>(pcls, ptxywh, grid_xy, anchor_wh,
                                                             inv_fsize, topKeys, selBox, selBoxOff,
                                                             selArea, selLabel, selIdb, selScore,
                                                             selValid);
  k_supp  <<<(kTopK / 16) * (kTopK / 16) / 8, 256, 0, stream>>>(selBoxOff, selArea, supp);
  k_nms_out<<<1,                           256, 0, stream>>>(supp, selValid, selBox, selLabel,
                                                             selIdb, selScore, out);
  (void)in_sizes; (void)n_in; (void)out_size; (void)ws_size;
}